// DefaultProjectorTorch_4020089389773
// MI455X (gfx1250) — compile-verified
//
#include <hip/hip_runtime.h>

// ---------------------------------------------------------------------------
// CDNA5 (gfx1250) projector: coeff[atom, n, lm] = W @ ( angs x (phi*rho) )
// Matrix engine path: V_WMMA_F32_16X16X4_F32, K accumulated over voxels.
// ---------------------------------------------------------------------------

typedef __attribute__((ext_vector_type(2))) float v2f;
typedef __attribute__((ext_vector_type(8))) float v8f;

#define GRID   180
#define NATOM  64
#define NRAD   8
#define NLM    16
#define RMAXI  13
#define BDIM   27                 // 2*RMAXI + 1
#define NVOX   (BDIM*BDIM*BDIM)   // 19683
#define RO     2.5f
#define CELL   0.2f               // L/G = 36/180
#define VCELLF 0.008f             // CELL^3
#define WAVES  8
#define NTILES ((NVOX + 31) / 32) // 616 -> 77 tiles per wave, uniform

__global__ __launch_bounds__(256)
void projector_wmma_kernel(const float* __restrict__ rho,
                           const float* __restrict__ pos,
                           const float* __restrict__ W,
                           float* __restrict__ out)
{
    // Per-wave staging for WMMA fragment assembly (wave-private -> no barriers).
    // s_psi is 16 columns wide: cols 8..15 are zeroed ONCE so the B fragment is
    // a plain strided LDS load with no per-WMMA cndmask zero-padding.
    __shared__ float s_angs[WAVES][32][NLM];   // 16 KB
    __shared__ float s_psi [WAVES][32][NLM];   // 16 KB
    // Cross-wave reduction + final 16x16 tile
    __shared__ float s_red [WAVES][256];       //  8 KB
    __shared__ float s_tile[NLM][16];          //  1 KB

    const int atom = blockIdx.x;
    const int tid  = threadIdx.x;
    const int lane = tid & 31;
    const int wave = tid >> 5;

    const float px = pos[atom * 3 + 0];
    const float py = pos[atom * 3 + 1];
    const float pz = pos[atom * 3 + 2];

    // cm = round(pos / a)  (rintf == RNE, matches jnp.round)
    const float cmx = rintf(px * 5.0f);
    const float cmy = rintf(py * 5.0f);
    const float cmz = rintf(pz * 5.0f);
    const int icx = (int)cmx, icy = (int)cmy, icz = (int)cmz;
    const float drx = px - CELL * cmx;
    const float dry = py - CELL * cmy;
    const float drz = pz - CELL * cmz;

    // zero the padding columns of this wave's psi rows (row == lane), once
    #pragma unroll
    for (int m = NRAD; m < NLM; ++m) s_psi[wave][lane][m] = 0.0f;
    __builtin_amdgcn_wave_barrier();

    v8f acc = {};   // 16(lm) x 16(n-padded) f32 accumulator tile

    const int m16 = lane & 15;          // A: lm row / B: n column
    const int kk  = (lane >> 4) * 2;    // K sub-slot (0 or 2) per ISA layout

    for (int tile = wave; tile < NTILES; tile += WAVES) {
        const int vox  = tile * 32 + lane;
        const int voxc = (vox < NVOX) ? vox : 0;

        // voxel -> (ix, iy, iz) in the 27^3 stencil
        const int ixo = voxc / (BDIM * BDIM);
        const int rem = voxc - ixo * (BDIM * BDIM);
        const int iyo = rem / BDIM;
        const int izo = rem - iyo * BDIM;

        // periodic wrap (cm in [0,180], off in [-13,13] -> one fold suffices)
        int gx = icx + ixo - RMAXI; gx += (gx < 0) ? GRID : 0; gx -= (gx >= GRID) ? GRID : 0;
        int gy = icy + iyo - RMAXI; gy += (gy < 0) ? GRID : 0; gy -= (gy >= GRID) ? GRID : 0;
        int gz = icz + izo - RMAXI; gz += (gz < 0) ? GRID : 0; gz -= (gz >= GRID) ? GRID : 0;

        const float srho = rho[(gx * GRID + gy) * GRID + gz];

        const float xs = (float)(ixo - RMAXI) * CELL - drx;
        const float ys = (float)(iyo - RMAXI) * CELL - dry;
        const float zs = (float)(izo - RMAXI) * CELL - drz;

        const float R2 = xs * xs + ys * ys + zs * zs;
        const float R  = sqrtf(R2);
        const float rinv = (R > 1e-12f) ? (1.0f / R) : 0.0f;
        const float u = xs * rinv;   // sin(th)cos(ph)
        const float v = ys * rinv;   // sin(th)sin(ph)
        const float w = zs * rinv;   // cos(th)

        // radial: phi_m = R^2 * max(RO-R,0)^(m+2); fold Vcell*rho in now,
        // defer the W-orthonormalization mix to after the reduction.
        float t = RO - R; t = (t > 0.0f) ? t : 0.0f;
        float p = R2 * t * t * (VCELLF)*srho;
        if (vox >= NVOX) p = 0.0f;   // zero-pad tail K columns via B=0
        #pragma unroll
        for (int m = 0; m < NRAD; ++m) { s_psi[wave][lane][m] = p; p *= t; }

        // real spherical harmonics as polynomials in (u,v,w)
        const float u2 = u * u, vv2 = v * v, w2 = w * w;
        float ang[NLM];
        ang[0]  = 0.28209479177387814f;
        ang[1]  = 0.4886025119029199f * v;
        ang[2]  = 0.4886025119029199f * w;
        ang[3]  = 0.4886025119029199f * u;
        ang[4]  = 1.0925484305920792f * u * v;
        ang[5]  = 1.0925484305920792f * v * w;
        ang[6]  = 0.31539156525252005f * (3.0f * w2 - 1.0f);
        ang[7]  = 1.0925484305920792f * u * w;
        ang[8]  = 0.5462742152960396f * (u2 - vv2);
        ang[9]  = 0.5900435899266435f * (3.0f * u2 - vv2) * v;
        ang[10] = 2.890611442640554f  * u * v * w;
        ang[11] = 0.4570457994644658f * v * (5.0f * w2 - 1.0f);
        ang[12] = 0.3731763325901154f * w * (5.0f * w2 - 3.0f);
        ang[13] = 0.4570457994644658f * u * (5.0f * w2 - 1.0f);
        ang[14] = 1.445305721320277f  * (u2 - vv2) * w;
        ang[15] = 0.5900435899266435f * u * (u2 - 3.0f * vv2);
        #pragma unroll
        for (int j = 0; j < NLM; ++j) s_angs[wave][lane][j] = ang[j];

        // wave-private LDS region: same-wave DS ordering makes store->load safe
        __builtin_amdgcn_wave_barrier();

        // 32 voxels = 8 x (K=4) WMMA steps
        #pragma unroll
        for (int k4 = 0; k4 < 8; ++k4) {
            const int v0 = k4 * 4 + kk;
            const int v1 = v0 + 1;
            v2f a, b;
            // A (16x4, M=lm): lane[0:15]->K{0,1}, lane[16:31]->K{2,3}
            a.x = s_angs[wave][v0][m16];
            a.y = s_angs[wave][v1][m16];
            // B (4x16, N=n): symmetric layout; cols n>=8 are pre-zeroed padding
            b.x = s_psi[wave][v0][m16];
            b.y = s_psi[wave][v1][m16];
            acc = __builtin_amdgcn_wmma_f32_16x16x4_f32(
                /*neg_a=*/false, a, /*neg_b=*/false, b,
                /*c_mod=*/(short)0, acc, /*reuse_a=*/false, /*reuse_b=*/false);
        }
        __builtin_amdgcn_wave_barrier();
    }

    // ---- cross-wave reduction of the 16x16 f32 tiles ----
    #pragma unroll
    for (int j = 0; j < 8; ++j) s_red[wave][j * 32 + lane] = acc[j];
    __syncthreads();

    {
        const int rlane = tid & 31;
        const int rv    = tid >> 5;
        float s = 0.0f;
        #pragma unroll
        for (int w8 = 0; w8 < WAVES; ++w8) s += s_red[w8][rv * 32 + rlane];
        // C/D layout: M(lm) = vgpr + 8*(lane>=16), N(n) = lane&15
        const int lm   = rv + ((rlane >= 16) ? 8 : 0);
        const int ncol = rlane & 15;
        s_tile[lm][ncol] = s;
    }
    __syncthreads();

    // ---- apply W orthonormalization (8x8) once per atom, masked store ----
    if (tid < 128) {
        const int n  = tid >> 4;
        const int lm = tid & 15;
        float s = 0.0f;
        #pragma unroll
        for (int m = 0; m < NRAD; ++m) s += s_tile[lm][m] * W[n * NRAD + m];
        out[atom * (NRAD * NLM) + n * NLM + lm] = s;
    }
}

extern "C" void kernel_launch(void* const* d_in, const int* in_sizes, int n_in,
                              void* d_out, int out_size, void* d_ws, size_t ws_size,
                              hipStream_t stream) {
    (void)in_sizes; (void)n_in; (void)out_size; (void)d_ws; (void)ws_size;
    const float* rho = (const float*)d_in[0];   // (180,180,180) f32
    const float* pos = (const float*)d_in[1];   // (64,3) f32
    const float* W   = (const float*)d_in[2];   // (8,8) f32
    float* out = (float*)d_out;                 // (64,128) f32
    projector_wmma_kernel<<<NATOM, 256, 0, stream>>>(rho, pos, W, out);
}